// FactorizedSpectralConv_31559419691632
// MI455X (gfx1250) — compile-verified
//
#include <hip/hip_runtime.h>
#include <hip/hip_bf16.h>

// FNO FactorizedSpectralConv for MI455X (gfx1250, wave32, WMMA).
// Pipeline: [K1] partial DFT over W (WMMA f32 GEMM, M=131072 K=512 N=32)
//           [K2] partial DFT over H (32 kx modes, LDS sincos table)
//           [K3] per-mode 32x32 complex channel contraction
//           [K4] inverse DFT over H
//           [K5] inverse DFT over W + bias (WMMA f32 GEMM, M=131072 K=32 N=512)
// Traffic ~560 MB total -> HBM-bound (~24us at 23.3 TB/s); fp32 WMMA keeps
// reference precision on the long (K=512) reductions.
//
// v2: twiddle matrices are stored in LDS pre-swizzled into the per-lane WMMA
// B-fragment layout (v2f per (ntile, kstep, lane)), so each B operand is one
// aligned ds_load_b64 directly into the register pair the WMMA reads --
// removes the v_mov shuffles seen in v1's inner loop and is LDS-bank-conflict
// free (32 lanes x 8B = all 64 banks once).

#define Bn   8
#define Ci   32
#define Co   32
#define Hn   512
#define Wn   512
#define HM   16            // ky modes
#define MX   32            // kx modes (16 top + 16 bottom)
#define Mrows (Bn*Ci*Hn)   // 131072
#define TWOPI_512 0.0122718463030851298f   // 2*pi/512

typedef float v2f __attribute__((ext_vector_type(2)));
typedef float v8f __attribute__((ext_vector_type(8)));

__device__ __forceinline__ v8f wmma_f32(v2f a, v2f b, v8f c) {
    // D = A(16x4 f32) * B(4x16 f32) + C(16x16 f32)
    return __builtin_amdgcn_wmma_f32_16x16x4_f32(
        /*neg_a=*/false, a, /*neg_b=*/false, b,
        /*c_mod=*/(short)0, c, /*reuse_a=*/false, /*reuse_b=*/false);
}

// DFT twiddle element for K1: column n<16 -> cos(2*pi*k*n/512),
// n>=16 -> -sin(2*pi*k*(n-16)/512).
__device__ __forceinline__ float fwd_tw(int k, int n) {
    const int ky = n & 15;
    float s, c;
    __sincosf((float)((k * ky) & 511) * TWOPI_512, &s, &c);
    return (n < 16) ? c : -s;
}

// ---------------------------------------------------------------------------
// K1: X1[row][n] = sum_w x[row][w] * T[w][n]      row = (b*Ci+i)*H + h
// B fragments pre-swizzled: Bt[ntile][k0/4][lane] = {T[k][col], T[k+1][col]}
// with k = k0 + 2*(lane>>4), col = ntile*16 + (lane&15).
// ---------------------------------------------------------------------------
__global__ void __launch_bounds__(128)
dft_w_kernel(const float* __restrict__ x, float* __restrict__ X1) {
    __shared__ v2f Bt[2][128][32];   // 64 KB swizzled twiddles
    const int t = threadIdx.x;
    for (int idx = t; idx < 2 * 128 * 32; idx += 128) {
        const int ln = idx & 31;
        const int k4 = (idx >> 5) & 127;
        const int nt = idx >> 12;
        const int col = nt * 16 + (ln & 15);
        const int k = k4 * 4 + 2 * (ln >> 4);
        v2f e; e.x = fwd_tw(k, col); e.y = fwd_tw(k + 1, col);
        Bt[nt][k4][ln] = e;
    }
    __syncthreads();

    const int wave = t >> 5, lane = t & 31;
    const int hl = lane >> 4, lm = lane & 15;
    const int ntiles = (Mrows / 16) / 4;   // 4 wave-tiles per block iter

    for (int tile = blockIdx.x; tile < ntiles; tile += gridDim.x) {
        const long m0 = ((long)tile * 4 + wave) * 16;
        const float* arow = x + (m0 + lm) * Wn + hl * 2;
        v8f acc0 = {}; v8f acc1 = {};
        #pragma unroll 4
        for (int k0 = 0; k0 < 512; k0 += 4) {
            v2f a = *(const v2f*)(arow + k0);   // (k, k+1), k = k0 + 2*hl
            v2f b0 = Bt[0][k0 >> 2][lane];
            v2f b1 = Bt[1][k0 >> 2][lane];
            acc0 = wmma_f32(a, b0, acc0);
            acc1 = wmma_f32(a, b1, acc1);
        }
        #pragma unroll
        for (int r = 0; r < 8; ++r) {
            const long row = m0 + r + 8 * hl;
            X1[row * 32 + lm]      = acc0[r];
            X1[row * 32 + 16 + lm] = acc1[r];
        }
    }
}

// ---------------------------------------------------------------------------
// K2: X2[b,i,kx,ky] = sum_h X1[b,i,h,ky] * e^{-2*pi*i*phys(kx)*h/512}
// ---------------------------------------------------------------------------
__global__ void __launch_bounds__(256)
dft_h_kernel(const float* __restrict__ X1, float* __restrict__ X2) {
    __shared__ float ct[512], st[512];
    for (int p = threadIdx.x; p < 512; p += 256) {
        float s, c; __sincosf((float)p * TWOPI_512, &s, &c);
        ct[p] = c; st[p] = s;
    }
    __syncthreads();

    const int gid = blockIdx.x * 256 + threadIdx.x;     // 131072 threads
    const int ky = gid & 15;
    const int kx = (gid >> 4) & 31;
    const int bi = gid >> 9;
    const int phys = (kx < 16) ? kx : (480 + kx);       // 496..511 for bottom
    const float* base = X1 + (long)bi * Hn * 32;
    float re = 0.f, im = 0.f;
    for (int h = 0; h < 512; ++h) {
        const float xr = base[h * 32 + ky];
        const float xi = base[h * 32 + 16 + ky];
        const int p = (phys * h) & 511;
        const float c = ct[p], s = st[p];
        re += xr * c + xi * s;     // * (cos - i sin)
        im += xi * c - xr * s;
    }
    const long o = ((long)bi * MX + kx) * HM + ky;
    X2[o * 2] = re; X2[o * 2 + 1] = im;
}

// ---------------------------------------------------------------------------
// K3: OFT[b,o,kx,ky] = sum_i X2[b,i,kx,ky] * W{1|2}[i,o,kxm,ky]   (complex)
// ---------------------------------------------------------------------------
__global__ void __launch_bounds__(256)
mode_mul_kernel(const float* __restrict__ X2,
                const float* __restrict__ w1r, const float* __restrict__ w1i,
                const float* __restrict__ w2r, const float* __restrict__ w2i,
                float* __restrict__ OFT) {
    const int gid = blockIdx.x * 256 + threadIdx.x;     // 131072 threads
    const int ky = gid & 15;
    const int kx = (gid >> 4) & 31;
    const int o  = (gid >> 9) & 31;
    const int b  = gid >> 14;
    const bool top = (kx < 16);
    const int kxm = top ? kx : kx - 16;
    const float* wr = top ? w1r : w2r;
    const float* wi = top ? w1i : w2i;
    float re = 0.f, im = 0.f;
    for (int i = 0; i < Ci; ++i) {
        const long xo = (((long)b * Ci + i) * MX + kx) * HM + ky;
        const float xr = X2[xo * 2], xi = X2[xo * 2 + 1];
        const long wo = (((long)i * Co + o) * HM + kxm) * HM + ky;
        const float a = wr[wo], bb = wi[wo];
        re += xr * a - xi * bb;
        im += xr * bb + xi * a;
    }
    const long oo = (((long)b * Co + o) * MX + kx) * HM + ky;
    OFT[oo * 2] = re; OFT[oo * 2 + 1] = im;
}

// ---------------------------------------------------------------------------
// K4: Y1c[row=(b*Co+o)*H+h][ky | 16+ky] = sum_kx OFT * e^{+2*pi*i*phys*h/512}
// ---------------------------------------------------------------------------
__global__ void __launch_bounds__(256)
idft_h_kernel(const float* __restrict__ OFT, float* __restrict__ Y1c) {
    __shared__ float ct[512], st[512];
    for (int p = threadIdx.x; p < 512; p += 256) {
        float s, c; __sincosf((float)p * TWOPI_512, &s, &c);
        ct[p] = c; st[p] = s;
    }
    __syncthreads();

    const int gid = blockIdx.x * 256 + threadIdx.x;     // 2,097,152 threads
    const int ky = gid & 15;
    const int h  = (gid >> 4) & 511;
    const int bo = gid >> 13;
    const float* base = OFT + (long)bo * MX * HM * 2;
    float re = 0.f, im = 0.f;
    for (int kx = 0; kx < MX; ++kx) {
        const int phys = (kx < 16) ? kx : (480 + kx);
        const int p = (phys * h) & 511;
        const float c = ct[p], s = st[p];
        const float orr = base[(kx * HM + ky) * 2];
        const float oi  = base[(kx * HM + ky) * 2 + 1];
        re += orr * c - oi * s;    // * (cos + i sin)
        im += orr * s + oi * c;
    }
    const long row = (long)bo * Hn + h;
    Y1c[row * 32 + ky]      = re;
    Y1c[row * 32 + 16 + ky] = im;
}

// iDFT matrix element for K5: row n<16 -> coef*cos(2*pi*n*w/512),
// n>=16 -> -coef*sin(...); coef folds Hermitian doubling and 1/(H*W).
__device__ __forceinline__ float inv_tw(int n, int w) {
    const float inv = 1.0f / (512.0f * 512.0f);
    const int ky = n & 15;
    float s, c;
    __sincosf((float)((ky * w) & 511) * TWOPI_512, &s, &c);
    const float coef = (ky == 0) ? inv : 2.0f * inv;
    return (n < 16) ? coef * c : ((ky == 0) ? 0.0f : -coef * s);
}

// ---------------------------------------------------------------------------
// K5: y[row][w] = sum_n Y1c[row][n] * Wm[n][w] + bias[o]
// B fragments pre-swizzled: Wt[ntile][k0/4][lane] = {Wm[k][col], Wm[k+1][col]}
// with k = k0 + 2*(lane>>4), col = ntile*16 + (lane&15);  32 ntiles cover N=512.
// ---------------------------------------------------------------------------
__global__ void __launch_bounds__(128)
idft_w_kernel(const float* __restrict__ Y1c, const float* __restrict__ bias,
              float* __restrict__ y) {
    __shared__ v2f Wt[32][8][32];   // 64 KB swizzled iDFT matrix
    const int t = threadIdx.x;
    for (int idx = t; idx < 32 * 8 * 32; idx += 128) {
        const int ln = idx & 31;
        const int k4 = (idx >> 5) & 7;
        const int nt = idx >> 8;
        const int col = nt * 16 + (ln & 15);
        const int k = k4 * 4 + 2 * (ln >> 4);
        v2f e; e.x = inv_tw(k, col); e.y = inv_tw(k + 1, col);
        Wt[nt][k4][ln] = e;
    }
    __syncthreads();

    const int wave = t >> 5, lane = t & 31;
    const int hl = lane >> 4, lm = lane & 15;

    for (int tile = blockIdx.x; tile < Mrows / 16; tile += gridDim.x) {
        const long m0 = (long)tile * 16;
        v2f areg[8];
        const float* arow = Y1c + (m0 + lm) * 32 + hl * 2;
        #pragma unroll
        for (int k0 = 0; k0 < 32; k0 += 4) areg[k0 >> 2] = *(const v2f*)(arow + k0);
        const float bsc = bias[(int)((m0 >> 9) & 31)];   // o = (row/512)%32

        for (int nt = 0; nt < 8; ++nt) {
            const int ng = wave * 8 + nt;            // global N-tile, 4 waves split N
            v8f acc = {};
            #pragma unroll
            for (int k0 = 0; k0 < 32; k0 += 4) {
                acc = wmma_f32(areg[k0 >> 2], Wt[ng][k0 >> 2][lane], acc);
            }
            const int nb = ng * 16;
            #pragma unroll
            for (int r = 0; r < 8; ++r) {
                const long row = m0 + r + 8 * hl;
                y[row * (long)Wn + nb + lm] = acc[r] + bsc;
            }
        }
    }
}

// ---------------------------------------------------------------------------
extern "C" void kernel_launch(void* const* d_in, const int* in_sizes, int n_in,
                              void* d_out, int out_size, void* d_ws, size_t ws_size,
                              hipStream_t stream) {
    const float* x    = (const float*)d_in[0];
    const float* w1r  = (const float*)d_in[1];
    const float* w1i  = (const float*)d_in[2];
    const float* w2r  = (const float*)d_in[3];
    const float* w2i  = (const float*)d_in[4];
    const float* bias = (const float*)d_in[5];
    float* y  = (float*)d_out;
    float* ws = (float*)d_ws;

    float* X1  = ws;                              // 131072*32       = 4,194,304 f
    float* X2  = X1  + (long)Mrows * 32;          // 8*32*32*16*2    =   262,144 f
    float* OFT = X2  + (long)Bn * Ci * MX * HM * 2;
    float* Y1c = OFT + (long)Bn * Co * MX * HM * 2;   // 131072*32

    dft_w_kernel   <<<1024, 128, 0, stream>>>(x, X1);
    dft_h_kernel   <<< 512, 256, 0, stream>>>(X1, X2);
    mode_mul_kernel<<< 512, 256, 0, stream>>>(X2, w1r, w1i, w2r, w2i, OFT);
    idft_h_kernel  <<<8192, 256, 0, stream>>>(OFT, Y1c);
    idft_w_kernel  <<< 512, 128, 0, stream>>>(Y1c, bias, y);
}